// GaussExpFalloffKernelGrad_83434034692737
// MI455X (gfx1250) — compile-verified
//
#include <hip/hip_runtime.h>

typedef __attribute__((ext_vector_type(2))) float v2f;
typedef __attribute__((ext_vector_type(8))) float v8f;

#define NPTS   65536   // 256*256 points per batch
#define NSRC   64      // vortices per batch

// One wave32 handles a 16-point tile against all 64 sources.
// sq_distance computed via V_WMMA_F32_16X16X4_F32 using the bilinear expansion
//   ||p - s||^2 = (-2y)*py + (-2x)*px + (y^2+x^2)*1 + 1*(py^2+px^2)
// A (16x4) = per-source rows, B (4x16) = per-point columns.
// D layout (ISA 7.12.2): lane L holds point N=L&15; VGPR r holds source
// M = r + (L>=16 ? 8 : 0). So each lane owns 1 point x 8 sources per tile.
__global__ __launch_bounds__(256)
void vortex_grad_kernel(const float* __restrict__ vf,   // (4,64,8): y,x,tau,sig,v,u,c,d
                        const float* __restrict__ pts,  // (4,256,256,2)
                        float* __restrict__ out)        // (4,256,256,4)
{
    const int lane   = threadIdx.x & 31;
    const int wave   = threadIdx.x >> 5;
    const int tile   = blockIdx.x * 8 + wave;   // 16-point tile id
    const int b      = tile >> 12;              // 4096 tiles per batch
    const int ptBase = (tile & 4095) << 4;
    const bool hi    = lane >= 16;
    const int  l15   = lane & 15;

    // This lane's point (point index = lane&15 within the tile).
    const float2 p  = *(const float2*)(pts + ((size_t)b * NPTS + ptBase + l15) * 2);
    const float  py = p.x, px = p.y;

    // B operand (points, K=4 rows x N=16 cols), 2 VGPRs:
    //   VGPR0: lanes0-15 -> K=0 row (py), lanes16-31 -> K=2 row (1.0)
    //   VGPR1: lanes0-15 -> K=1 row (px), lanes16-31 -> K=3 row (py^2+px^2)
    v2f Bm;
    Bm.x = hi ? 1.0f             : py;
    Bm.y = hi ? (py*py + px*px)  : px;

    float acc0 = 0.f, acc1 = 0.f, acc2 = 0.f, acc3 = 0.f;   // gvy, gvx, guy, gux

    const float* fb = vf + (size_t)b * NSRC * 8;

    #pragma unroll
    for (int t = 0; t < 4; ++t) {
        // A operand (sources, M=16 rows x K=4): lane supplies row M = lane&15.
        //   VGPR0: lanes0-15 -> K=0 (-2y), lanes16-31 -> K=2 (y^2+x^2)
        //   VGPR1: lanes0-15 -> K=1 (-2x), lanes16-31 -> K=3 (1.0)
        const int   sA = t * 16 + l15;
        const float yA = fb[sA * 8 + 0];
        const float xA = fb[sA * 8 + 1];
        v2f Am;
        Am.x = hi ? (yA*yA + xA*xA) : (-2.0f * yA);
        Am.y = hi ? 1.0f            : (-2.0f * xA);

        v8f csq = {};
        csq = __builtin_amdgcn_wmma_f32_16x16x4_f32(
                  /*neg_a=*/false, Am, /*neg_b=*/false, Bm,
                  /*c_mod=*/(short)0, csq, /*reuse_a=*/false, /*reuse_b=*/false);

        // Per-pair nonlinear math: this lane's point vs its 8 sources.
        const int sBase = t * 16 + (hi ? 8 : 0);
        #pragma unroll
        for (int r = 0; r < 8; ++r) {
            const float4 f0 = *(const float4*)(fb + (size_t)(sBase + r) * 8);     // y,x,tau,sig
            const float4 f1 = *(const float4*)(fb + (size_t)(sBase + r) * 8 + 4); // v,u,c,d
            const float y = f0.x, x = f0.y, tau = f0.z, sig = f0.w;
            const float c = f1.z, d = f1.w;

            const float sq = csq[r];                 // from WMMA
            const float d1 = py - y;
            const float d2 = px - x;

            const float sig2      = sig * sig;
            const float inv_dsig2 = __builtin_amdgcn_rcpf(d * sig2);   // 1/(d*sig^2)
            const float inv_sig2  = d * inv_dsig2;                     // 1/sig^2

            const float exp1  = __expf(-10.0f * sq * inv_dsig2);       // DT=1
            const float gauss = __expf(-sq * inv_sig2);
            const float ce1   = c * exp1;

            const float tden  = sq + ce1;
            const float rin   = __builtin_amdgcn_rsqf(tden);           // 1/sqrt(.)
            const float denom = tden * rin;                            // sqrt(.)
            const float expP  = __expf(-ce1) * rin;
            const float fall  = gauss * expP;

            // grad_kernel_r * r  (r^2 = sq folds away the sqrt for r)
            const float gkr = 2.0f * fall * sq * (1.0f - ce1 * inv_dsig2)
                              * (0.5f * rin - denom);
            const float m   = gkr + fall;

            const float i1  = __builtin_amdgcn_rcpf(d1);
            const float i2  = __builtin_amdgcn_rcpf(d2);
            const float r12 = d1 * i2;      // d1/d2
            const float r21 = d2 * i1;      // d2/d1

            const float guy = -tau * (m + fall);
            acc0 += tau * (r21 * m - r12 * fall);   // grad_v_y
            acc1 -= guy;                            // grad_v_x =  tau*(m+fall)
            acc2 += guy;                            // grad_u_y
            acc3 -= tau * (r12 * m - r21 * fall);   // grad_u_x
        }
    }

    // Lane L and L+16 hold the same point (sources split 8/8): pair-reduce.
    acc0 += __shfl_xor(acc0, 16, 32);
    acc1 += __shfl_xor(acc1, 16, 32);
    acc2 += __shfl_xor(acc2, 16, 32);
    acc3 += __shfl_xor(acc3, 16, 32);

    if (!hi) {
        float4 o = make_float4(acc0, acc1, acc2, acc3);
        *(float4*)(out + ((size_t)b * NPTS + ptBase + l15) * 4) = o;
    }
}

extern "C" void kernel_launch(void* const* d_in, const int* in_sizes, int n_in,
                              void* d_out, int out_size, void* d_ws, size_t ws_size,
                              hipStream_t stream) {
    const float* vf  = (const float*)d_in[0];   // (4,64,8)
    const float* pts = (const float*)d_in[1];   // (4,256,256,2)
    float* out       = (float*)d_out;           // (4,256,256,4)

    const int npoints = in_sizes[1] / 2;        // 262144
    const int tiles   = npoints / 16;           // 16384 (16-point tiles)
    const int blocks  = (tiles + 7) / 8;        // 8 waves per 256-thread block

    vortex_grad_kernel<<<blocks, 256, 0, stream>>>(vf, pts, out);
}